// DocREModel_79139067396612
// MI455X (gfx1250) — compile-verified
//
#include <hip/hip_runtime.h>
#include <math.h>

// ---------------- problem constants (from reference) ----------------
#define N_DOC   4
#define C_LEN   1024
#define D_EMB   768
#define H_HEADS 12
#define E_ENT   42
#define M_MENT  8
#define P_PAIR  1024
#define EMB     768
#define BS      64
#define NB      12          // EMB / BS
#define NCLS    99
#define POS_OFF 1
#define NROWS   (N_DOC * P_PAIR)   // 4096

typedef float v2f __attribute__((ext_vector_type(2)));
typedef float v8f __attribute__((ext_vector_type(8)));
typedef unsigned int u32x4 __attribute__((ext_vector_type(4)));
typedef int i32x4 __attribute__((ext_vector_type(4)));
typedef int i32x8 __attribute__((ext_vector_type(8)));

// Force GLOBAL (addrspace 1) loads: avoids FLAT path (LOADcnt+DScnt, LDS arb).
typedef __attribute__((address_space(1))) const float* gcfp;
typedef __attribute__((address_space(1))) const v2f*  gcv2p;
static __device__ __forceinline__ float gld(const float* p) {
    return *(gcfp)(unsigned long long)(size_t)p;
}
static __device__ __forceinline__ v2f gld2(const float* p) {   // 8B-aligned pair
    return *(gcv2p)(unsigned long long)(size_t)p;
}

// fp32 WMMA: D(16x16) = A(16x4) * B(4x16) + C   -> v_wmma_f32_16x16x4_f32
static __device__ __forceinline__ v8f wmma_f32x4(v2f a, v2f b, v8f c) {
    return __builtin_amdgcn_wmma_f32_16x16x4_f32(
        /*neg_a=*/false, a, /*neg_b=*/false, b,
        /*c_mod=*/(short)0, c, /*reuse_a=*/false, /*reuse_b=*/false);
}

// ---- Tensor Data Mover: 2D tile (tile_rows x row_elems fp32) global->LDS ----
// D# built per CDNA5 ISA 8.3/8.4. pad_enable: +4 DWORDs every 256 DWORDs,
// so a 768-float row lands with LDS pitch 780 floats (bank-conflict-free
// column walks), in-row column c maps to c + 4*(c>>8).
// This toolchain's builtin takes 6 args (g0,g1,g2,g3,g4,cpol).
static __device__ __forceinline__ void tdm_load_2d(unsigned lds_byte_off,
                                                   const void* gptr,
                                                   int tile_rows, int row_elems,
                                                   int row_stride_elems) {
    unsigned long long ga = (unsigned long long)(size_t)gptr;
    u32x4 g0;
    g0[0] = 1u;                                        // count=1, user mode
    g0[1] = lds_byte_off;                              // lds_addr (bytes)
    g0[2] = (unsigned)(ga & 0xFFFFFFFFu);              // global_addr[31:0]
    g0[3] = (unsigned)((ga >> 32) & 0x1FFFFFFu) | (2u << 30);  // [56:32] | type=2
    i32x8 g1;
    g1[0] = (2 << 16) | (1 << 20) | (7 << 22) | (3 << 25);
            // data_size=4B | pad_enable | pad_interval=256DW | pad_amount=4DW
    g1[1] = (row_elems & 0xFFFF) << 16;                // tensor_dim0 lo16
    g1[2] = ((row_elems >> 16) & 0xFFFF) | ((tile_rows & 0xFFFF) << 16);
            // tensor_dim0 hi16 | tensor_dim1 lo16
    g1[3] = ((tile_rows >> 16) & 0xFFFF) | ((row_elems & 0xFFFF) << 16);
            // tensor_dim1 hi16 | tile_dim0
    g1[4] = (tile_rows & 0xFFFF);                      // tile_dim1 | tile_dim2=0
    g1[5] = row_stride_elems;                          // tensor_dim0_stride lo32
    g1[6] = 0;
    g1[7] = 0;
    i32x4 z4 = {0, 0, 0, 0};
    i32x8 z8 = {0, 0, 0, 0, 0, 0, 0, 0};
    __builtin_amdgcn_tensor_load_to_lds(g0, g1, z4, z4, z8, 0);
}

// ---------------- kernel A: entity embedding (masked logsumexp over M) ----
__global__ void k_entity_emb(const float* __restrict__ seq,
                             const int* __restrict__ pos,
                             const int* __restrict__ mask,
                             float* __restrict__ e_emb) {
    int be  = blockIdx.x;                // doc*E + e
    int doc = be / E_ENT;
    const int* p  = pos  + (size_t)be * M_MENT;
    const int* mk = mask + (size_t)be * M_MENT;
    int pm[M_MENT]; int vm[M_MENT];
    #pragma unroll
    for (int m = 0; m < M_MENT; ++m) { pm[m] = p[m] + POS_OFF; vm[m] = mk[m]; }
    const float* sdoc = seq + (size_t)doc * C_LEN * D_EMB;
    for (int j = threadIdx.x; j < D_EMB; j += blockDim.x) {
        float mx = -INFINITY;
        #pragma unroll
        for (int m = 0; m < M_MENT; ++m)
            if (vm[m] > 0) mx = fmaxf(mx, gld(sdoc + (size_t)pm[m] * D_EMB + j));
        float s = 0.f;
        #pragma unroll
        for (int m = 0; m < M_MENT; ++m)
            if (vm[m] > 0) s += expf(gld(sdoc + (size_t)pm[m] * D_EMB + j) - mx);
        e_emb[(size_t)be * D_EMB + j] = mx + logf(s);
    }
}

// ---------------- kernel B: entity attention (masked mean over mentions) ----
// e_att layout: [doc][e][head][c]
__global__ void k_entity_att(const float* __restrict__ att,
                             const int* __restrict__ pos,
                             const int* __restrict__ mask,
                             float* __restrict__ e_att) {
    int b    = blockIdx.x;               // (doc*E + e)*H + head
    int head = b % H_HEADS;
    int de   = b / H_HEADS;              // doc*E + e
    int doc  = de / E_ENT;
    const int* p  = pos  + (size_t)de * M_MENT;
    const int* mk = mask + (size_t)de * M_MENT;
    int pm[M_MENT]; int vm[M_MENT]; int cnt = 0;
    #pragma unroll
    for (int m = 0; m < M_MENT; ++m) { pm[m] = p[m] + POS_OFF; vm[m] = mk[m]; cnt += (mk[m] > 0); }
    float inv = 1.0f / (float)cnt;
    const float* adoc = att + ((size_t)doc * H_HEADS + head) * (size_t)C_LEN * C_LEN;
    float* out = e_att + ((size_t)de * H_HEADS + head) * C_LEN;
    for (int k = threadIdx.x; k < C_LEN; k += blockDim.x) {
        float s = 0.f;
        #pragma unroll
        for (int m = 0; m < M_MENT; ++m)
            if (vm[m] > 0) s += gld(adoc + (size_t)pm[m] * C_LEN + k);
        out[k] = s * inv;
    }
}

// ---------------- kernel C: ht_att (head*tail product, mean over heads, L1-normalized)
__global__ void k_ht_att(const float* __restrict__ e_att,
                         const int* __restrict__ hts,
                         float* __restrict__ ht_att) {
    int b   = blockIdx.x;                // doc*P + r
    int doc = b / P_PAIR;
    int h_idx = hts[(size_t)b * 2 + 0];
    int t_idx = hts[(size_t)b * 2 + 1];
    const float* ha = e_att + (size_t)(doc * E_ENT + h_idx) * H_HEADS * C_LEN;
    const float* ta = e_att + (size_t)(doc * E_ENT + t_idx) * H_HEADS * C_LEN;
    __shared__ float red[8];
    float vals[C_LEN / 256];
    float local = 0.f;
    #pragma unroll
    for (int kk = 0; kk < C_LEN / 256; ++kk) {
        int k = threadIdx.x + kk * 256;
        float s = 0.f;
        #pragma unroll
        for (int hh = 0; hh < H_HEADS; ++hh)
            s += gld(ha + hh * C_LEN + k) * gld(ta + hh * C_LEN + k);
        s *= (1.0f / H_HEADS);
        vals[kk] = s;
        local += s;
    }
    #pragma unroll
    for (int off = 16; off >= 1; off >>= 1) local += __shfl_xor(local, off, 32);
    int wid = threadIdx.x >> 5;
    if ((threadIdx.x & 31) == 0) red[wid] = local;
    __syncthreads();
    if (threadIdx.x == 0) { float t = 0.f; for (int i = 0; i < 8; ++i) t += red[i]; red[0] = t; }
    __syncthreads();
    float inv = 1.0f / (red[0] + 1e-5f);
    float* out = ht_att + (size_t)b * C_LEN;
    #pragma unroll
    for (int kk = 0; kk < C_LEN / 256; ++kk) {
        int k = threadIdx.x + kk * 256;
        out[k] = vals[kk] * inv;
    }
}

// ---------------- transpose helper (Wh/Wt -> K-major for coalesced B frags) ----
__global__ void k_transpose(const float* __restrict__ W, float* __restrict__ WT,
                            int rows, int cols) {
    int idx = blockIdx.x * blockDim.x + threadIdx.x;
    if (idx < rows * cols) {
        int rr = idx / cols, cc = idx % cols;
        WT[(size_t)cc * rows + rr] = W[idx];
    }
}

// ---------------- kernel D: rs = ht_att(P x c) @ seq(c x d) per doc ----------
// One wave computes a 64(M) x 32(N) C-tile: 4x2 v8f accumulators, 8 WMMAs per
// K-step. Single base pointers + immediate offsets keep everything on the
// GLOBAL path (LOADcnt only, no flat/DScnt coupling).
__global__ void k_rs_gemm(const float* __restrict__ ht_att,
                          const float* __restrict__ seq,
                          float* __restrict__ rs) {
    int doc  = blockIdx.z;
    int r0   = blockIdx.x * 64;
    int o0   = blockIdx.y * 32;
    int lane = threadIdx.x;
    int half = lane >> 4, l = lane & 15;
    const float* A = ht_att + (size_t)doc * P_PAIR * C_LEN;        // [r][k]
    const float* B = seq    + (size_t)doc * C_LEN  * D_EMB;        // [k][o]
    v8f acc[4][2];
    #pragma unroll
    for (int ms = 0; ms < 4; ++ms)
        #pragma unroll
        for (int ns = 0; ns < 2; ++ns) acc[ms][ns] = (v8f){};
    const float* aBase = A + (size_t)(r0 + l) * C_LEN + half * 2;
    const float* bBase = B + (size_t)(half * 2) * D_EMB + (o0 + l);
    for (int k0 = 0; k0 < C_LEN; k0 += 4) {
        v2f a[4];
        #pragma unroll
        for (int ms = 0; ms < 4; ++ms)
            a[ms] = gld2(aBase + (size_t)(ms * 16) * C_LEN + k0);
        v2f b[2];
        b[0].x = gld(bBase);      b[0].y = gld(bBase + D_EMB);
        b[1].x = gld(bBase + 16); b[1].y = gld(bBase + D_EMB + 16);
        bBase += 4 * D_EMB;
        #pragma unroll
        for (int ms = 0; ms < 4; ++ms)
            #pragma unroll
            for (int ns = 0; ns < 2; ++ns)
                acc[ms][ns] = wmma_f32x4(a[ms], b[ns], acc[ms][ns]);
    }
    float* out = rs + (size_t)(doc * P_PAIR + r0) * D_EMB + o0;
    #pragma unroll
    for (int ms = 0; ms < 4; ++ms)
        #pragma unroll
        for (int ns = 0; ns < 2; ++ns)
            #pragma unroll
            for (int v = 0; v < 8; ++v) {
                int m = ms * 16 + v + half * 8;
                out[(size_t)m * D_EMB + ns * 16 + l] = acc[ms][ns][v];
            }
}

// ---------------- kernel E: tanh([gathered e_emb | rs] @ W^T + b) -------------
// Same 64x32 register blocking; K split at 768 so the concat gather is two
// clean loops (e_emb rows for k<768, rs rows for k>=768).
__global__ void k_head_tail(const float* __restrict__ e_emb,
                            const float* __restrict__ rs,
                            const int* __restrict__ hts,
                            const float* __restrict__ WT,   // [2d][EMB], K-major
                            const float* __restrict__ bias,
                            float* __restrict__ out, int col) {
    int r0   = blockIdx.x * 64;
    int o0   = blockIdx.y * 32;
    int lane = threadIdx.x;
    int half = lane >> 4, l = lane & 15;
    const float* eRow[4];                 // gathered -> explicit global loads
    #pragma unroll
    for (int ms = 0; ms < 4; ++ms) {
        int row = r0 + ms * 16 + l;
        int doc = row / P_PAIR, pr = row % P_PAIR;
        int ent = hts[((size_t)doc * P_PAIR + pr) * 2 + col];
        eRow[ms] = e_emb + (size_t)(doc * E_ENT + ent) * D_EMB + half * 2;
    }
    const float* rBase = rs + (size_t)(r0 + l) * D_EMB + half * 2;
    v8f acc[4][2];
    #pragma unroll
    for (int ms = 0; ms < 4; ++ms)
        #pragma unroll
        for (int ns = 0; ns < 2; ++ns) acc[ms][ns] = (v8f){};
    const float* bPtr = WT + (size_t)(half * 2) * EMB + (o0 + l);
    for (int k0 = 0; k0 < D_EMB; k0 += 4) {              // first half: e_emb
        v2f a[4];
        #pragma unroll
        for (int ms = 0; ms < 4; ++ms)
            a[ms] = gld2(eRow[ms] + k0);
        v2f b[2];
        b[0].x = gld(bPtr);      b[0].y = gld(bPtr + EMB);
        b[1].x = gld(bPtr + 16); b[1].y = gld(bPtr + EMB + 16);
        bPtr += 4 * EMB;
        #pragma unroll
        for (int ms = 0; ms < 4; ++ms)
            #pragma unroll
            for (int ns = 0; ns < 2; ++ns)
                acc[ms][ns] = wmma_f32x4(a[ms], b[ns], acc[ms][ns]);
    }
    for (int k0 = 0; k0 < D_EMB; k0 += 4) {              // second half: rs
        v2f a[4];
        #pragma unroll
        for (int ms = 0; ms < 4; ++ms)
            a[ms] = gld2(rBase + (size_t)(ms * 16) * D_EMB + k0);
        v2f b[2];
        b[0].x = gld(bPtr);      b[0].y = gld(bPtr + EMB);
        b[1].x = gld(bPtr + 16); b[1].y = gld(bPtr + EMB + 16);
        bPtr += 4 * EMB;
        #pragma unroll
        for (int ms = 0; ms < 4; ++ms)
            #pragma unroll
            for (int ns = 0; ns < 2; ++ns)
                acc[ms][ns] = wmma_f32x4(a[ms], b[ns], acc[ms][ns]);
    }
    float* op = out + (size_t)r0 * EMB + o0;
    #pragma unroll
    for (int ns = 0; ns < 2; ++ns) {
        float bo = bias[o0 + ns * 16 + l];
        #pragma unroll
        for (int ms = 0; ms < 4; ++ms)
            #pragma unroll
            for (int v = 0; v < 8; ++v) {
                int m = ms * 16 + v + half * 8;
                op[(size_t)m * EMB + ns * 16 + l] = tanhf(acc[ms][ns][v] + bo);
            }
    }
}

// ---------------- kernel F: fused bilinear-block classifier --------------------
// logits[r,k] = sum_b  hs[r,b,:](1x64) @ Wb[k,b](64x64) @ ts[r,b,:](64x1)
// hs/ts tiles staged into LDS by the Tensor Data Mover (pad 4DW/256DW ->
// pitch 780 floats, conflict-free column walks). Never materializes bl.
#define LDSPITCH 780
static __device__ __forceinline__ int padc(int c) { return c + ((c >> 8) << 2); }

__global__ void k_logits(const float* __restrict__ hs,
                         const float* __restrict__ ts,
                         const float* __restrict__ Wb,
                         float* __restrict__ logits) {
    extern __shared__ float lds[];
    float* U = lds;                    // [16][LDSPITCH]
    float* V = lds + 16 * LDSPITCH;    // [16][LDSPITCH]
    int r0 = blockIdx.x * 16;
    if ((threadIdx.x >> 5) == 0) {     // wave 0 drives the TDM (EXEC ignored)
        tdm_load_2d(0u, hs + (size_t)r0 * EMB, 16, EMB, EMB);
        tdm_load_2d((unsigned)(16 * LDSPITCH * sizeof(float)),
                    ts + (size_t)r0 * EMB, 16, EMB, EMB);
        __builtin_amdgcn_s_wait_tensorcnt(0);
    }
    __syncthreads();
    int wave = threadIdx.x >> 5;
    int lane = threadIdx.x & 31;
    int half = lane >> 4, l = lane & 15;
    for (int k = wave; k < NCLS; k += 8) {
        if (k + 8 < NCLS)              // global_prefetch_b8 next Wb slab
            __builtin_prefetch(Wb + ((size_t)(k + 8) * NB) * (BS * BS), 0, 1);
        float part[8];
        #pragma unroll
        for (int v = 0; v < 8; ++v) part[v] = 0.f;
        for (int b = 0; b < NB; ++b) {
            const float* Wk = Wb + ((size_t)k * NB + b) * (BS * BS);  // [i][j]
            #pragma unroll
            for (int nsub = 0; nsub < 4; ++nsub) {
                int n0 = nsub * 16;
                v8f acc = (v8f){};
                for (int kk = 0; kk < BS; kk += 4) {
                    int colA = b * BS + kk + half * 2;
                    v2f a;
                    a.x = U[l * LDSPITCH + padc(colA)];
                    a.y = U[l * LDSPITCH + padc(colA) + 1];
                    v2f bb;
                    bb.x = gld(Wk + (kk + half * 2) * BS + n0 + l);
                    bb.y = gld(Wk + (kk + half * 2 + 1) * BS + n0 + l);
                    acc = wmma_f32x4(a, bb, acc);
                }
                #pragma unroll
                for (int v = 0; v < 8; ++v) {
                    int m = v + half * 8;                 // D row
                    part[v] += acc[v] * V[m * LDSPITCH + padc(b * BS + n0 + l)];
                }
            }
        }
        #pragma unroll
        for (int v = 0; v < 8; ++v) {                     // reduce over 16 lanes/half
            float x = part[v];
            x += __shfl_xor(x, 1, 32);
            x += __shfl_xor(x, 2, 32);
            x += __shfl_xor(x, 4, 32);
            x += __shfl_xor(x, 8, 32);
            part[v] = x;
        }
        if (l == 0) {
            #pragma unroll
            for (int v = 0; v < 8; ++v) {
                int m = v + half * 8;
                logits[(size_t)(r0 + m) * NCLS + k] = part[v];
            }
        }
    }
}

// ---------------- launch ------------------------------------------------------
extern "C" void kernel_launch(void* const* d_in, const int* in_sizes, int n_in,
                              void* d_out, int out_size, void* d_ws, size_t ws_size,
                              hipStream_t stream) {
    (void)in_sizes; (void)n_in; (void)out_size; (void)ws_size;
    const float* seq  = (const float*)d_in[0];
    const float* att  = (const float*)d_in[1];
    const float* Wh   = (const float*)d_in[2];
    const float* bh   = (const float*)d_in[3];
    const float* Wt   = (const float*)d_in[4];
    const float* bt   = (const float*)d_in[5];
    const float* Wb   = (const float*)d_in[6];
    const int*   pos  = (const int*)d_in[7];
    const int*   mask = (const int*)d_in[8];
    const int*   hts  = (const int*)d_in[9];
    float* logits = (float*)d_out;

    float* w = (float*)d_ws;
    float* e_emb  = w; w += (size_t)N_DOC * E_ENT * D_EMB;            // 129024
    float* e_att  = w; w += (size_t)N_DOC * E_ENT * H_HEADS * C_LEN;  // 2064384
    float* ht_att = w; w += (size_t)N_DOC * P_PAIR * C_LEN;           // 4194304
    float* rs     = w; w += (size_t)NROWS * D_EMB;                    // 3145728
    float* hs     = w; w += (size_t)NROWS * EMB;                      // 3145728
    float* ts     = w; w += (size_t)NROWS * EMB;                      // 3145728
    float* WhT    = w; w += (size_t)2 * D_EMB * EMB;                  // 1179648
    float* WtT    = w; w += (size_t)2 * D_EMB * EMB;                  // 1179648

    k_entity_emb<<<N_DOC * E_ENT, 256, 0, stream>>>(seq, pos, mask, e_emb);
    k_entity_att<<<N_DOC * E_ENT * H_HEADS, 256, 0, stream>>>(att, pos, mask, e_att);
    k_ht_att<<<N_DOC * P_PAIR, 256, 0, stream>>>(e_att, hts, ht_att);

    int tn = EMB * 2 * D_EMB;
    k_transpose<<<(tn + 255) / 256, 256, 0, stream>>>(Wh, WhT, EMB, 2 * D_EMB);
    k_transpose<<<(tn + 255) / 256, 256, 0, stream>>>(Wt, WtT, EMB, 2 * D_EMB);

    k_rs_gemm<<<dim3(P_PAIR / 64, D_EMB / 32, N_DOC), 32, 0, stream>>>(ht_att, seq, rs);

    k_head_tail<<<dim3(NROWS / 64, EMB / 32), 32, 0, stream>>>(e_emb, rs, hts, WhT, bh, hs, 0);
    k_head_tail<<<dim3(NROWS / 64, EMB / 32), 32, 0, stream>>>(e_emb, rs, hts, WtT, bt, ts, 1);

    size_t ldsF = (size_t)2 * 16 * LDSPITCH * sizeof(float);   // ~99.8 KB dynamic LDS
    k_logits<<<NROWS / 16, 256, ldsF, stream>>>(hs, ts, Wb, logits);
}